// ContextMatching_14594298872548
// MI455X (gfx1250) — compile-verified
//
#include <hip/hip_runtime.h>
#include <math.h>

// Problem constants (match reference).
#define BB 16
#define T1 1024
#define T2 1024
#define CC 1024

typedef __attribute__((ext_vector_type(2))) float v2f;
typedef __attribute__((ext_vector_type(8))) float v8f;

// ---------------------------------------------------------------------------
// Kernel 1: p2[b,j] = <s2[b,j,:], w2>   (w2 = w[C:2C])
// wave-per-row, 8 waves/block. Reads s2 (64 MB) once at full HBM BW.
// ---------------------------------------------------------------------------
__global__ void cm_p2_kernel(const float* __restrict__ s2,
                             const float* __restrict__ w,
                             float* __restrict__ p2ws) {
    const int tid  = threadIdx.x;
    const int wave = tid >> 5;
    const int lane = tid & 31;
    const int row  = blockIdx.x * 8 + wave;          // flat over B*T2
    const float* srow = s2 + (size_t)row * CC;
    const float* w2   = w + CC;

    float sum = 0.0f;
#pragma unroll
    for (int k = 0; k < CC; k += 128) {              // 32 lanes * float4 = 128 floats
        const float4 sv = *(const float4*)(srow + k + lane * 4);
        const float4 wv = *(const float4*)(w2   + k + lane * 4);
        sum += sv.x * wv.x + sv.y * wv.y + sv.z * wv.z + sv.w * wv.w;
    }
    // wave32 reduction
#pragma unroll
    for (int off = 16; off > 0; off >>= 1)
        sum += __shfl_down(sum, off, 32);
    if (lane == 0) p2ws[row] = sum;
}

// ---------------------------------------------------------------------------
// Kernel 2: per-batch masked softmax over j (in-place on p2ws).
// One block per batch, 256 threads, LDS tree reductions. l2[b] >= 1 so the
// denominator is always > 0 for valid rows.
// ---------------------------------------------------------------------------
__global__ void cm_softmax_kernel(float* __restrict__ p2ws,
                                  const int* __restrict__ l2) {
    __shared__ float red[256];
    const int b   = blockIdx.x;
    const int tid = threadIdx.x;
    const int L   = l2[b];
    float* pb = p2ws + b * T2;

    float v[4];
    float m = -INFINITY;
#pragma unroll
    for (int r = 0; r < 4; ++r) {
        const int j = tid + r * 256;
        v[r] = pb[j];
        if (j < L) m = fmaxf(m, v[r]);
    }
    red[tid] = m;
    __syncthreads();
#pragma unroll
    for (int s = 128; s > 0; s >>= 1) {
        if (tid < s) red[tid] = fmaxf(red[tid], red[tid + s]);
        __syncthreads();
    }
    m = red[0];
    __syncthreads();

    float e[4];
    float sum = 0.0f;
#pragma unroll
    for (int r = 0; r < 4; ++r) {
        const int j = tid + r * 256;
        e[r] = (j < L) ? expf(v[r] - m) : 0.0f;
        sum += e[r];
    }
    red[tid] = sum;
    __syncthreads();
#pragma unroll
    for (int s = 128; s > 0; s >>= 1) {
        if (tid < s) red[tid] += red[tid + s];
        __syncthreads();
    }
    const float inv = 1.0f / red[0];
#pragma unroll
    for (int r = 0; r < 4; ++r)
        pb[tid + r * 256] = e[r] * inv;
}

// ---------------------------------------------------------------------------
// Kernel 3: ctx[b,c] = sum_j a[b,j] * s2[b,j,c]  via V_WMMA_F32_16X16X4_F32.
// Exact fp32 accumulate (RNE) to match the fp32 reference einsum.
// 8 blocks per batch; each of the 8 waves owns one 16-wide c-tile and runs
// the K loop over T2 in steps of 4. The `a` row is broadcast across all 16
// M-rows (M-waste irrelevant: this whole pass is ~33 MFLOP, L2-resident).
//
// Operand layouts (ISA 7.12.2, wave32):
//   A 16x4 f32 : lanes 0-15 -> {K=0,K=1}, lanes 16-31 -> {K=2,K=3}
//   B 4x16 f32 : VGPR r: row (r + 2*half) striped over lanes (N = lane&15)
//   D 16x16    : VGPR0 lanes 0-15 = row M=0, N = lane
// EXEC is all-1s at every WMMA (no divergence before the store).
// ---------------------------------------------------------------------------
__global__ void cm_ctx_wmma_kernel(const float* __restrict__ s2,
                                   const float* __restrict__ aws,
                                   float* __restrict__ ctxws) {
    __shared__ float a_lds[T2];
    const int tid  = threadIdx.x;
    const int b    = blockIdx.x >> 3;                // 8 blocks per batch
#pragma unroll
    for (int r = 0; r < 4; ++r)
        a_lds[tid + r * 256] = aws[b * T2 + tid + r * 256];
    __syncthreads();

    const int wave = tid >> 5;
    const int lane = tid & 31;
    const int half = lane >> 4;                      // 0: K 0..1, 1: K 2..3
    const int n    = lane & 15;
    const int c0   = ((blockIdx.x & 7) * 8 + wave) * 16;
    const float* s2b = s2 + (size_t)b * T2 * CC;

    v8f acc = {};
#pragma unroll 4
    for (int j0 = 0; j0 < T2; j0 += 4) {
        const int k = j0 + 2 * half;
        v2f A; A.x = a_lds[k];                       // broadcast row of weights
        A.y = a_lds[k + 1];
        v2f Bm; Bm.x = s2b[(size_t)k * CC + c0 + n];
        Bm.y = s2b[(size_t)(k + 1) * CC + c0 + n];
        acc = __builtin_amdgcn_wmma_f32_16x16x4_f32(
            /*neg_a=*/false, A, /*neg_b=*/false, Bm,
            /*c_mod=*/(short)0, acc, /*reuse_a=*/false, /*reuse_b=*/false);
    }
    // Row M=0 of D = ctx[c0 .. c0+15], held in acc[0] of lanes 0..15.
    if (lane < 16) ctxws[b * CC + c0 + lane] = acc[0];
}

// ---------------------------------------------------------------------------
// Kernel 4: out[b,i,:] = (i < l1[b]) ? ctx[b,:] : 0.  The bandwidth pass:
// 64 MB of float4 stores (~2.7 us at 23.3 TB/s); ctx rows are L2-resident.
// One block per output row, fully overwrites d_out.
// ---------------------------------------------------------------------------
__global__ void cm_bcast_kernel(const float* __restrict__ ctxws,
                                const int* __restrict__ l1,
                                float* __restrict__ out) {
    const int row = blockIdx.x;                      // b*T1 + i
    const int b   = row >> 10;
    const int i   = row & (T1 - 1);
    const int tid = threadIdx.x;

    float4 v = make_float4(0.0f, 0.0f, 0.0f, 0.0f);
    if (i < l1[b])
        v = ((const float4*)(ctxws + b * CC))[tid];
    ((float4*)out)[(size_t)row * (CC / 4) + tid] = v;
}

// ---------------------------------------------------------------------------
// Inputs (setup_inputs order): 0=s1 (dead), 1=l1, 2=s2, 3=l2, 4=w.
// ws layout: [0, 64KB) p2 -> softmax weights (in-place); [64KB, 128KB) ctx.
// ---------------------------------------------------------------------------
extern "C" void kernel_launch(void* const* d_in, const int* in_sizes, int n_in,
                              void* d_out, int out_size, void* d_ws, size_t ws_size,
                              hipStream_t stream) {
    (void)in_sizes; (void)n_in; (void)out_size; (void)ws_size;
    const int*   l1 = (const int*)d_in[1];
    const float* s2 = (const float*)d_in[2];
    const int*   l2 = (const int*)d_in[3];
    const float* w  = (const float*)d_in[4];
    float* out   = (float*)d_out;
    float* p2ws  = (float*)d_ws;              // B*T2 floats = 64 KB
    float* ctxws = p2ws + BB * T2;            // B*C  floats = 64 KB

    cm_p2_kernel     <<<BB * T2 / 8, 256, 0, stream>>>(s2, w, p2ws);
    cm_softmax_kernel<<<BB,          256, 0, stream>>>(p2ws, l2);
    cm_ctx_wmma_kernel<<<BB * 8,     256, 0, stream>>>(s2, p2ws, ctxws);
    cm_bcast_kernel  <<<BB * T1,     256, 0, stream>>>(ctxws, l1, out);
}